// TrafficGNN_15453292331591
// MI455X (gfx1250) — compile-verified
//
#include <hip/hip_runtime.h>

// ---------------------------------------------------------------------------
// Temporal GNN (GATx2 + GRU + FC) for MI455X (gfx1250, wave32).
//   - Dense GEMMs (x@W^T) use V_WMMA_F32_16X16X4_F32 (fp32-native matrix op),
//     templated on (K,C) so fragment loads are branch-free b64 loads.
//   - Edge softmax-aggregation uses L2-resident atomics (whole working set
//     < 192MB L2, so HBM is never the limiter; edge passes are L2-bound).
// ---------------------------------------------------------------------------

#define FIN_DIM  3
#define HID_DIM  32
#define NHEADS   2
#define NEG_SLOPE 0.2f

typedef float v2f __attribute__((ext_vector_type(2)));
typedef float v8f __attribute__((ext_vector_type(8)));

// ---- ordered-key mapping so float max can use unsigned atomicMax ----------
__device__ __forceinline__ unsigned f32_key(float f) {
  int i = __float_as_int(f);
  return (i >= 0) ? ((unsigned)i | 0x80000000u) : (unsigned)(~i);
}
__device__ __forceinline__ float f32_unkey(unsigned u) {
  return (u & 0x80000000u) ? __int_as_float((int)(u & 0x7fffffffu))
                           : __int_as_float(~(int)u);
}

__device__ __forceinline__ float lrelu(float v) { return v > 0.f ? v : NEG_SLOPE * v; }
__device__ __forceinline__ float sigmoidf(float v) { return 1.f / (1.f + expf(-v)); }

// ---------------------------------------------------------------------------
// zero fill
__global__ void zero_kernel(float* p, long long n) {
  long long i = (long long)blockIdx.x * blockDim.x + threadIdx.x;
  if (i < n) p[i] = 0.f;
}

// ---------------------------------------------------------------------------
// Y[Nrows,C] = X[Nrows,K] @ W[C,K]^T (+ bias[C]) via V_WMMA_F32_16X16X4_F32.
// One wave per 16x16 output tile; K padded to multiple of 4.
// Fragment layouts (CDNA5 ISA):
//   A 16x4 : lane M = lane&15, VGPR v holds K = (lane>>4)*2 + v
//   B 4x16 : lane N = lane&15, VGPR v holds row K = (lane>>4)*2 + v
//   C/D    : VGPR r holds row M = r + 8*(lane>>4), col N = lane&15
template <int K, int C>
__global__ void wmma_gemm_kernel(const float* __restrict__ X,
                                 const float* __restrict__ W,
                                 const float* __restrict__ bias,
                                 float* __restrict__ Y, int Nrows) {
  constexpr int KP = (K + 3) & ~3;
  constexpr int ctiles = C / 16;
  int wave = (int)((blockIdx.x * blockDim.x + threadIdx.x) >> 5);
  int lane = threadIdx.x & 31;
  int ntiles = (Nrows >> 4) * ctiles;
  if (wave >= ntiles) return;            // wave-uniform exit (EXEC stays full)
  int n0 = (wave / ctiles) * 16;
  int c0 = (wave % ctiles) * 16;
  int lh = lane >> 4;
  int m  = lane & 15;
  const float* Xr = X + (size_t)(n0 + m) * K;   // A row for this lane
  const float* Wr = W + (size_t)(c0 + m) * K;   // B column (= W row) for lane

  v8f acc = {};
#pragma unroll
  for (int kb = 0; kb < KP; kb += 4) {
    int k0 = kb + lh * 2;
    v2f a, b;
    if constexpr ((K & 3) == 0) {
      // rows are 8B-aligned at even k -> single b64 loads, no predication
      a = *(const v2f*)(Xr + k0);
      b = *(const v2f*)(Wr + k0);
    } else {
      // K==3 path: clamp address in-range, mask value (v_cndmask, no branch)
      int   ka0 = (k0     < K) ? k0     : K - 1;
      int   ka1 = (k0 + 1 < K) ? k0 + 1 : K - 1;
      float m0  = (k0     < K) ? 1.f : 0.f;
      float m1  = (k0 + 1 < K) ? 1.f : 0.f;
      a.x = Xr[ka0] * m0;  a.y = Xr[ka1] * m1;
      b.x = Wr[ka0] * m0;  b.y = Wr[ka1] * m1;
    }
    acc = __builtin_amdgcn_wmma_f32_16x16x4_f32(false, a, false, b,
                                                (short)0, acc, false, false);
  }

  float bs = bias ? bias[c0 + m] : 0.f;
#pragma unroll
  for (int r = 0; r < 8; ++r)
    Y[(size_t)(n0 + r + 8 * lh) * C + c0 + m] = acc[r] + bs;
}

// ---------------------------------------------------------------------------
// per-node attention scalars: asrc[n,h]=<h64[n,h,:],a_src[h,:]>, same for dst
__global__ void attn_sums_kernel(const float* __restrict__ h64,
                                 const float* __restrict__ a_src,
                                 const float* __restrict__ a_dst,
                                 float* __restrict__ asrc,
                                 float* __restrict__ adst, int Nn) {
  int i = blockIdx.x * blockDim.x + threadIdx.x;    // (node, head)
  if (i >= Nn * NHEADS) return;
  int n = i >> 1, head = i & 1;
  const float* hp = h64 + (size_t)n * (NHEADS * HID_DIM) + head * HID_DIM;
  const float* as = a_src + head * HID_DIM;
  const float* ad = a_dst + head * HID_DIM;
  float ss = 0.f, sd = 0.f;
#pragma unroll
  for (int k = 0; k < HID_DIM; ++k) { float hv = hp[k]; ss += hv * as[k]; sd += hv * ad[k]; }
  asrc[i] = ss; adst[i] = sd;
}

// ---------------------------------------------------------------------------
// pass A: per-destination segment max of e = lrelu(asrc[src]+adst[dst])
__global__ void edge_max_kernel(const float* __restrict__ asrc,
                                const float* __restrict__ adst,
                                const int* __restrict__ ei,
                                unsigned* __restrict__ mkey, int E, int Nn) {
  long long i = (long long)blockIdx.x * blockDim.x + threadIdx.x;
  long long total = (long long)(E + Nn) * NHEADS;
  if (i >= total) return;
  int head = (int)(i & 1);
  long long e = i >> 1;
  int s, d;
  if (e < E) { s = ei[e]; d = ei[(long long)E + e]; } else { s = d = (int)(e - E); }
  float v = lrelu(asrc[(size_t)s * 2 + head] + adst[(size_t)d * 2 + head]);
  atomicMax(&mkey[(size_t)d * 2 + head], f32_key(v));
}

// ---------------------------------------------------------------------------
// pass B: unnormalized softmax aggregation. One lane per (edge, head, k):
//   ex = exp(e - m[dst]); agg[dst,head,k] += h[src,head,k]*ex; denom += ex
__global__ void edge_agg_kernel(const float* __restrict__ h64,
                                const float* __restrict__ asrc,
                                const float* __restrict__ adst,
                                const unsigned* __restrict__ mkey,
                                const int* __restrict__ ei,
                                float* __restrict__ agg,
                                float* __restrict__ denom, int E, int Nn) {
  long long gid = (long long)blockIdx.x * blockDim.x + threadIdx.x;
  long long total = (long long)(E + Nn) * NHEADS * HID_DIM;
  if (gid >= total) return;
  int k = (int)(gid & (HID_DIM - 1));
  long long p = gid >> 5;
  int head = (int)(p & 1);
  long long e = p >> 1;
  int s, d;
  if (e < E) { s = ei[e]; d = ei[(long long)E + e]; } else { s = d = (int)(e - E); }
  float v  = lrelu(asrc[(size_t)s * 2 + head] + adst[(size_t)d * 2 + head]);
  float mv = f32_unkey(mkey[(size_t)d * 2 + head]);
  float ex = expf(v - mv);
  float val = h64[(size_t)s * 64 + head * HID_DIM + k] * ex;
  atomicAdd(&agg[(size_t)d * 64 + head * HID_DIM + k], val);
  if (k == 0) atomicAdd(&denom[(size_t)d * 2 + head], ex);
}

// ---------------------------------------------------------------------------
// pass C: normalize, mean over heads, +bias, ReLU
__global__ void gat_finalize_kernel(const float* __restrict__ agg,
                                    const float* __restrict__ denom,
                                    const float* __restrict__ bias,
                                    float* __restrict__ g, int Nn) {
  int i = blockIdx.x * blockDim.x + threadIdx.x;   // (node, k)
  if (i >= Nn * HID_DIM) return;
  int n = i >> 5, k = i & (HID_DIM - 1);
  float d0 = denom[(size_t)n * 2 + 0] + 1e-16f;
  float d1 = denom[(size_t)n * 2 + 1] + 1e-16f;
  float v = 0.5f * (agg[(size_t)n * 64 + k] / d0 +
                    agg[(size_t)n * 64 + HID_DIM + k] / d1) + bias[k];
  g[i] = v > 0.f ? v : 0.f;
}

// ---------------------------------------------------------------------------
// GRU cell elementwise (torch gate order r,z,n); in-place hstate update
__global__ void gru_gate_kernel(const float* __restrict__ gi,
                                const float* __restrict__ gh,
                                float* __restrict__ hstate, int Nn) {
  int i = blockIdx.x * blockDim.x + threadIdx.x;   // (node, k)
  if (i >= Nn * HID_DIM) return;
  int n = i >> 5, k = i & (HID_DIM - 1);
  const float* Gi = gi + (size_t)n * 96;
  const float* Gh = gh + (size_t)n * 96;
  float r = sigmoidf(Gi[k]      + Gh[k]);
  float z = sigmoidf(Gi[32 + k] + Gh[32 + k]);
  float c = tanhf(Gi[64 + k] + r * Gh[64 + k]);
  float ho = hstate[i];
  hstate[i] = (1.f - z) * c + z * ho;
}

// ---------------------------------------------------------------------------
// final Linear(32 -> 1)
__global__ void fc_kernel(const float* __restrict__ hstate,
                          const float* __restrict__ fc_w,
                          const float* __restrict__ fc_b,
                          float* __restrict__ out, int Nn) {
  int n = blockIdx.x * blockDim.x + threadIdx.x;
  if (n >= Nn) return;
  float s = fc_b[0];
#pragma unroll
  for (int k = 0; k < HID_DIM; ++k) s += hstate[(size_t)n * HID_DIM + k] * fc_w[k];
  out[n] = s;
}

// ---------------------------------------------------------------------------
static inline int cdiv(long long a, long long b) { return (int)((a + b - 1) / b); }

extern "C" void kernel_launch(void* const* d_in, const int* in_sizes, int n_in,
                              void* d_out, int out_size, void* d_ws, size_t ws_size,
                              hipStream_t stream) {
  (void)n_in; (void)ws_size;
  const float* x_seq  = (const float*)d_in[0];
  const float* W1     = (const float*)d_in[1];
  const float* a1_src = (const float*)d_in[2];
  const float* a1_dst = (const float*)d_in[3];
  const float* b1     = (const float*)d_in[4];
  const float* W2     = (const float*)d_in[5];
  const float* a2_src = (const float*)d_in[6];
  const float* a2_dst = (const float*)d_in[7];
  const float* b2     = (const float*)d_in[8];
  const float* w_ih   = (const float*)d_in[9];
  const float* w_hh   = (const float*)d_in[10];
  const float* b_ih   = (const float*)d_in[11];
  const float* b_hh   = (const float*)d_in[12];
  const float* fc_w   = (const float*)d_in[13];
  const float* fc_b   = (const float*)d_in[14];
  const int*   ei     = (const int*)d_in[15];   // int32 (JAX x64 disabled)
  float* out = (float*)d_out;

  const int N = out_size;                       // 50000 (divisible by 16)
  const int E = in_sizes[15] / 2;               // 800000
  const int T = in_sizes[0] / (N * FIN_DIM);    // 8

  // ---- workspace carve (all fp32-sized; ~392*N floats = 78.4 MB) ----
  float* p = (float*)d_ws;
  float*    h64    = p; p += (size_t)N * 64;
  float*    agg    = p; p += (size_t)N * 64;    // agg|denom|mkey contiguous
  float*    denom  = p; p += (size_t)N * 2;     //   -> single zero pass (N*68)
  unsigned* mkey   = (unsigned*)p; p += (size_t)N * 2;
  float*    asrc   = p; p += (size_t)N * 2;
  float*    adst   = p; p += (size_t)N * 2;
  float*    g      = p; p += (size_t)N * 32;
  float*    gi     = p; p += (size_t)N * 96;
  float*    gh     = p; p += (size_t)N * 96;
  float*    hstate = p; p += (size_t)N * 32;

  const int BLK = 256;
  const long long EE2  = (long long)(E + N) * NHEADS;
  const long long EE64 = (long long)(E + N) * NHEADS * HID_DIM;
  const int tiles64 = (N / 16) * (64 / 16);
  const int tiles96 = (N / 16) * (96 / 16);

  zero_kernel<<<cdiv((long long)N * 32, BLK), BLK, 0, stream>>>(hstate, (long long)N * 32);

  for (int t = 0; t < T; ++t) {
    const float* xt = x_seq + (size_t)t * N * FIN_DIM;

    // ---------- GAT layer 1 ----------
    wmma_gemm_kernel<FIN_DIM, 64><<<cdiv(tiles64, 8), BLK, 0, stream>>>(xt, W1, nullptr, h64, N);
    attn_sums_kernel<<<cdiv((long long)N * 2, BLK), BLK, 0, stream>>>(h64, a1_src, a1_dst, asrc, adst, N);
    zero_kernel<<<cdiv((long long)N * 68, BLK), BLK, 0, stream>>>(agg, (long long)N * 68);
    edge_max_kernel<<<cdiv(EE2, BLK), BLK, 0, stream>>>(asrc, adst, ei, mkey, E, N);
    edge_agg_kernel<<<cdiv(EE64, BLK), BLK, 0, stream>>>(h64, asrc, adst, mkey, ei, agg, denom, E, N);
    gat_finalize_kernel<<<cdiv((long long)N * 32, BLK), BLK, 0, stream>>>(agg, denom, b1, g, N);

    // ---------- GAT layer 2 ----------
    wmma_gemm_kernel<HID_DIM, 64><<<cdiv(tiles64, 8), BLK, 0, stream>>>(g, W2, nullptr, h64, N);
    attn_sums_kernel<<<cdiv((long long)N * 2, BLK), BLK, 0, stream>>>(h64, a2_src, a2_dst, asrc, adst, N);
    zero_kernel<<<cdiv((long long)N * 68, BLK), BLK, 0, stream>>>(agg, (long long)N * 68);
    edge_max_kernel<<<cdiv(EE2, BLK), BLK, 0, stream>>>(asrc, adst, ei, mkey, E, N);
    edge_agg_kernel<<<cdiv(EE64, BLK), BLK, 0, stream>>>(h64, asrc, adst, mkey, ei, agg, denom, E, N);
    gat_finalize_kernel<<<cdiv((long long)N * 32, BLK), BLK, 0, stream>>>(agg, denom, b2, g, N);

    // ---------- GRU cell ----------
    wmma_gemm_kernel<HID_DIM, 96><<<cdiv(tiles96, 8), BLK, 0, stream>>>(g,      w_ih, b_ih, gi, N);
    wmma_gemm_kernel<HID_DIM, 96><<<cdiv(tiles96, 8), BLK, 0, stream>>>(hstate, w_hh, b_hh, gh, N);
    gru_gate_kernel<<<cdiv((long long)N * 32, BLK), BLK, 0, stream>>>(gi, gh, hstate, N);
  }

  fc_kernel<<<cdiv(N, BLK), BLK, 0, stream>>>(hstate, fc_w, fc_b, out, N);
}